// QuantumNeuralNetwork_29867202576593
// MI455X (gfx1250) — compile-verified
//
#include <hip/hip_runtime.h>

// ---------------------------------------------------------------------------
// 16-qubit, 2-layer BasicEntangler QNN, batch 32.
//   Circuit reduces to:  psi3 = RX(W1) * P * RX(x+W0) |0>,  out = |P psi3|^2 @ signs
//   where P (the CNOT ring) is linear over GF(2) and folded into indexing/signs.
// Phase 1: per (sample, state-quarter) workgroup; analytic generation of
//          P*RX(x+W0)|0> with the bit15/bit14 RX(W1) gates folded in; then 14
//          in-LDS RX gate sweeps; writes probabilities p[b][u] (8 MB in d_ws).
//          Generation uses a shared 13-factor base product across the four
//          (t15,t14) fold variants (they only perturb s-bits 13..15).
// Phase 2: out = p @ signs via V_WMMA_F32_16X16X4_F32, signs = +/-1 generated
//          from hoisted bit parities in-register. 128 waves, partials to d_ws.
// Phase 3: deterministic fixed-order reduction of partials -> d_out (32x16).
// ---------------------------------------------------------------------------

#define NQ      16
#define QBITS   14
#define QN      (1 << QBITS)     // 16384 amplitudes per quarter (128 KB LDS)
#define NSTATE  (1 << NQ)        // 65536
#define BATCH   32
#define THREADS 1024
#define KCHUNKS 64
#define CHUNK   (NSTATE / KCHUNKS)   // 1024

typedef __attribute__((ext_vector_type(2))) float v2f;
typedef __attribute__((ext_vector_type(8))) float v8f;

__global__ __launch_bounds__(THREADS) void qsim_phase1(
    const float* __restrict__ x, const float* __restrict__ wts, float* __restrict__ p) {
  __shared__ float sre[QN];
  __shared__ float sim_[QN];
  const int wg  = blockIdx.x;
  const int b   = wg >> 2;
  const int hi  = wg & 3;          // (t15<<1)|t14 for this quarter
  const int tid = threadIdx.x;

  // Per-bit-position gate factors. bit j <-> wire w = 15-j.
  float cA[16], sA[16], cC[16], sC[16];
#pragma unroll
  for (int j = 0; j < 16; ++j) {
    const int w   = 15 - j;
    const float thA = 0.5f * (x[b * 16 + w] + wts[w]);  // embedding RX + layer0 RX merge
    const float thC = 0.5f * wts[16 + w];               // layer1 RX
    cA[j] = __builtin_cosf(thA); sA[j] = __builtin_sinf(thA);
    cC[j] = __builtin_cosf(thC); sC[j] = __builtin_sinf(thC);
  }

  // ---- analytic generation of psi2 = P * RX(x+W0)|0>, with the bit15 & bit14
  //      RX(W1) gates folded in.
  // psi2[t] = psi1[P^{-1} t]; P^{-1} bits: s[j] = t[j]^t[j+1] (j=0..13),
  // s[14] = t14^t0^t15, s[15] = t0^t15.  Factor per s-bit j: c[j] (bit 0) or
  // (-i)*s[j] (bit 1) -> magnitude product times (-i)^popcount.
  // For fixed l, the four fold variants t = (g<<14)|l differ only in s-bits
  // 13,14,15, so the product/popcount over s-bits 0..12 is shared.
  const float c15 = cC[15], s15 = sC[15], c14 = cC[14], s14 = sC[14];
  for (int l = tid; l < QN; l += THREADS) {
    const unsigned ul    = (unsigned)l;
    const unsigned low13 = (ul ^ (ul >> 1)) & 0x1FFFu;   // s-bits 0..12 (from t0..t13)
    const unsigned t13   = (ul >> 13) & 1u;
    const unsigned t0    = ul & 1u;
    const int   k_sh = __popc(low13);
    float m_sh = 1.0f;
#pragma unroll
    for (int j = 0; j < 13; ++j) m_sh *= ((low13 >> j) & 1u) ? sA[j] : cA[j];

    float r[4], q[4];
#pragma unroll
    for (int g = 0; g < 4; ++g) {
      const unsigned t14 = (unsigned)g & 1u;
      const unsigned t15 = ((unsigned)g >> 1) & 1u;
      const unsigned b13 = t13 ^ t14;
      const unsigned b14 = t14 ^ t0 ^ t15;
      const unsigned b15 = t0 ^ t15;
      const float m = m_sh * (b13 ? sA[13] : cA[13])
                           * (b14 ? sA[14] : cA[14])
                           * (b15 ? sA[15] : cA[15]);
      const int k = k_sh + (int)(b13 + b14 + b15);
      // (-i)^k : 0->(m,0) 1->(0,-m) 2->(-m,0) 3->(0,m)
      r[g] = (k & 1) ? 0.0f : ((k & 2) ? -m : m);
      q[g] = (k & 1) ? ((k & 2) ? m : -m) : 0.0f;
    }
    // RX(W1) on bit15: pairs (g, g^2).  a0' = c a0 - i s a1, etc.
#pragma unroll
    for (int g0 = 0; g0 < 2; ++g0) {
      const int g1 = g0 | 2;
      const float r0 = r[g0], i0 = q[g0], r1 = r[g1], i1 = q[g1];
      r[g0] = c15 * r0 + s15 * i1;  q[g0] = c15 * i0 - s15 * r1;
      r[g1] = c15 * r1 + s15 * i0;  q[g1] = c15 * i1 - s15 * r0;
    }
    // RX(W1) on bit14: pairs (g, g^1).
#pragma unroll
    for (int g0 = 0; g0 < 4; g0 += 2) {
      const int g1 = g0 | 1;
      const float r0 = r[g0], i0 = q[g0], r1 = r[g1], i1 = q[g1];
      r[g0] = c14 * r0 + s14 * i1;  q[g0] = c14 * i0 - s14 * r1;
      r[g1] = c14 * r1 + s14 * i0;  q[g1] = c14 * i1 - s14 * r0;
    }
    sre[l] = r[hi];
    sim_[l] = q[hi];
  }
  __syncthreads();

  // ---- remaining 14 RX(W1) gates on local bits 0..13, all in LDS ----
  // (access pattern is bank-conflict-free for all qb with 64 banks / 32 lanes)
#pragma unroll
  for (int qb = 0; qb < QBITS; ++qb) {
    const float c = cC[qb], s = sC[qb];
    const int half = 1 << qb;
    for (int pi = tid; pi < QN / 2; pi += THREADS) {
      const int l0 = ((pi >> qb) << (qb + 1)) | (pi & (half - 1));
      const int l1 = l0 | half;
      const float r0 = sre[l0], i0 = sim_[l0], r1 = sre[l1], i1 = sim_[l1];
      sre[l0] = c * r0 + s * i1;  sim_[l0] = c * i0 - s * r1;
      sre[l1] = c * r1 + s * i0;  sim_[l1] = c * i1 - s * r0;
    }
    __syncthreads();
  }

  // ---- probabilities out (coalesced) ----
  float* pb = p + (size_t)b * NSTATE + (size_t)hi * QN;
  for (int l = tid; l < QN; l += THREADS) {
    const float r = sre[l], i2 = sim_[l];
    pb[l] = r * r + i2 * i2;
  }
}

// out(32x16) = p(32x65536) @ signs(65536x16) via f32 WMMA.
// Sign for column w at state index u: 1 - 2*parity(u & M_w), with
// M_0 = 0x7FFF (bits 0..14) and M_w = (0xFFFF << (15-w)) & 0xFFFF for w>=1
// (final CNOT-ring permutation folded into the sign masks).
// Since kb is 4-aligned and u = kb | c, parity(u&M) = parity(kb&M)^parity(c&M):
// the c-dependent parts are hoisted to per-lane constants.
__global__ __launch_bounds__(32) void qsim_phase2(const float* __restrict__ p,
                                                  float* __restrict__ partial) {
  const int blk  = blockIdx.x;        // 0..127
  const int rb   = blk >> 6;          // sample row-block (0..1)
  const int ch   = blk & (KCHUNKS - 1);
  const int lane = threadIdx.x;       // 0..31
  const int m    = lane & 15;         // A row (sample within block) / B column
  const int hih  = lane >> 4;         // lane half -> K offset 2*hih

  const unsigned M = (m == 0) ? 0x7FFFu : ((0xFFFFu << (15 - m)) & 0xFFFFu);
  // c0 = 2*hih, c1 = 2*hih+1 :  hoisted parity constants
  const unsigned pc0 = (unsigned)__popc((unsigned)(2 * hih) & M) & 1u;
  const float    e01 = (M & 1u) ? -1.0f : 1.0f;   // parity delta between c0 and c0|1
  const float    s0f = pc0 ? -1.0f : 1.0f;

  const float* prow = p + (size_t)(rb * 16 + m) * NSTATE;

  v8f acc = {};
  const int kbeg = ch * CHUNK;
#pragma unroll 4
  for (int kb = kbeg; kb < kbeg + CHUNK; kb += 4) {
    const int u0 = kb + 2 * hih;
    // A tile (16x4): lane m, K = 2*hih + {0,1}
    v2f a;
    a.x = prow[u0];
    a.y = prow[u0 + 1];
    // B tile (4x16): row K = 2*hih + {0,1}, col n=m; +/-1 from parity
    const unsigned pk = (unsigned)__popc((unsigned)kb & M) & 1u;
    v2f bm;
    bm.x = pk ? -s0f : s0f;
    bm.y = bm.x * e01;
    acc = __builtin_amdgcn_wmma_f32_16x16x4_f32(false, a, false, bm,
                                                (short)0, acc, false, false);
  }
  // Store partial C tile: acc[r] at this lane = C[r + 8*hih][m]
  float* dst = partial + (size_t)blk * 256 + lane;
#pragma unroll
  for (int r = 0; r < 8; ++r) dst[r * 32] = acc[r];
}

__global__ void qsim_phase3(const float* __restrict__ partial, float* __restrict__ out) {
  const int t = blockIdx.x * blockDim.x + threadIdx.x;
  if (t >= BATCH * NQ) return;
  const int b = t >> 4, w = t & 15;
  const int rb = b >> 4, mc = b & 15;
  const int r = mc & 7, lane = w + ((mc >> 3) << 4);
  const float* base = partial + (size_t)(rb * KCHUNKS) * 256 + r * 32 + lane;
  float s = 0.0f;
#pragma unroll 8
  for (int ch = 0; ch < KCHUNKS; ++ch) s += base[(size_t)ch * 256];
  out[t] = s;
}

extern "C" void kernel_launch(void* const* d_in, const int* in_sizes, int n_in,
                              void* d_out, int out_size, void* d_ws, size_t ws_size,
                              hipStream_t stream) {
  const float* x   = (const float*)d_in[0];   // (32,16)
  const float* wts = (const float*)d_in[1];   // (2,16)
  float* out = (float*)d_out;                 // (32,16) float32
  float* p       = (float*)d_ws;                        // 32*65536 floats = 8 MB
  float* partial = p + (size_t)BATCH * NSTATE;          // 128*256 floats

  qsim_phase1<<<BATCH * 4, THREADS, 0, stream>>>(x, wts, p);
  qsim_phase2<<<2 * KCHUNKS, 32, 0, stream>>>(p, partial);
  qsim_phase3<<<(BATCH * NQ + 255) / 256, 256, 0, stream>>>(partial, out);
}